// MultiPillarCounter_712964571563
// MI455X (gfx1250) — compile-verified
//
#include <hip/hip_runtime.h>
#include <stdint.h>

// ---------------- static problem constants (match reference's static grid_sizes) ----
// grids: round(204.8/ps) = 4096, 2731, 2048, 1365, 1024, 683
#define NS 6
#define ACC_WORDS 32u             // [0..5]=counts [6..11]=mins [12..17]=maxs, rest pad
#define MASK_WORDS 994006u        // exact sum of ceil(g*g/32)
#define CHUNK_WORDS 4096u         // 16 KB LDS staging buffer (x2 for double buffer)
#define NCHUNKS 243u              // ceil(994006/4096)
#define MASK_WORDS_PAD (NCHUNKS * CHUNK_WORDS)     // 995328, zero-padded tail
#define TOTAL_WS_WORDS (ACC_WORDS + MASK_WORDS_PAD)

__device__ __forceinline__ unsigned grid_of(int s) {
  const unsigned g[NS] = {4096u, 2731u, 2048u, 1365u, 1024u, 683u};
  return g[s];
}
// cumulative word offsets of each size's bitmask region
__device__ __forceinline__ unsigned woff_of(int s) {
  const unsigned o[NS] = {0u, 524288u, 757362u, 888434u, 946660u, 979428u};
  return o[s];
}

#if __has_builtin(__builtin_amdgcn_s_wait_asynccnt)
#define WAIT_ASYNC(n) __builtin_amdgcn_s_wait_asynccnt(n)
#else
#define WAIT_ASYNC(n) asm volatile("s_wait_asynccnt %0" ::"i"(n) : "memory")
#endif

// async copy 16 bytes global -> LDS (tracked by ASYNCcnt)
static __device__ __forceinline__ void async_g2l_b128(unsigned lds_byte_off,
                                                      const unsigned* gsrc) {
  asm volatile("global_load_async_to_lds_b128 %0, %1, off"
               :: "v"(lds_byte_off), "v"(gsrc)
               : "memory");
}

// issue one full 4096-word chunk: 256 threads x 4 async b128 each
static __device__ __forceinline__ void issue_chunk(const unsigned* __restrict__ mask,
                                                   unsigned base, unsigned* bufrow,
                                                   unsigned tid) {
#pragma unroll
  for (unsigned k = 0; k < 4u; ++k) {
    unsigned w = tid * 4u + k * 1024u;
    async_g2l_b128((unsigned)(uintptr_t)&bufrow[w], mask + base + w);
  }
}

// ---------------- kernel 1: zero bitmask, init accumulators ------------------------
__global__ void mpc_init(unsigned* __restrict__ ws) {
  unsigned i = blockIdx.x * blockDim.x + threadIdx.x;
  unsigned stride = gridDim.x * blockDim.x;
  for (; i < TOTAL_WS_WORDS; i += stride) {
    unsigned v = 0u;
    if (i >= 6u && i < 12u)       v = 0x7FFFFFFFu;  // mins = INT_MAX
    else if (i >= 12u && i < 18u) v = 0x80000000u;  // maxs = INT_MIN
    ws[i] = v;
  }
}

// ---------------- kernel 2: scatter bits + min/max reduction -----------------------
__global__ void mpc_scatter(const unsigned long long* __restrict__ pts64,
                            const float* __restrict__ ps,     // (6,2)
                            const float* __restrict__ pcmin,  // (2,)
                            unsigned* __restrict__ ws, int npts) {
  const float pc0 = pcmin[0], pc1 = pcmin[1];
  float psx[NS], psy[NS];
#pragma unroll
  for (int s = 0; s < NS; ++s) { psx[s] = ps[2 * s]; psy[s] = ps[2 * s + 1]; }

  int lmin[NS], lmax[NS];
#pragma unroll
  for (int s = 0; s < NS; ++s) { lmin[s] = 0x7FFFFFFF; lmax[s] = (int)0x80000000; }

  int i = blockIdx.x * blockDim.x + threadIdx.x;
  const int stride = gridDim.x * blockDim.x;
  for (; i < npts; i += stride) {
    if (i + stride < npts) __builtin_prefetch(&pts64[i + stride], 0, 1);  // global_prefetch_b8
    // non-temporal streaming read: keep L2 reserved for the atomic bitmask working set
    unsigned long long raw = __builtin_nontemporal_load(pts64 + i);
    float2 p;
    __builtin_memcpy(&p, &raw, sizeof(p));
#pragma unroll
    for (int s = 0; s < NS; ++s) {
      const int g = (int)grid_of(s);
      // bit-match jnp.floor((p - pc_min) / pillar_size) in f32 (IEEE divide)
      int ix = (int)floorf((p.x - pc0) / psx[s]);
      int iy = (int)floorf((p.y - pc1) / psy[s]);
      lmin[s] = min(lmin[s], ix);
      lmax[s] = max(lmax[s], ix);
      // clamp like jnp scatter's default OOB clamping (no-op for in-range points)
      int cx = min(max(ix, 0), g - 1);
      int cy = min(max(iy, 0), g - 1);
      unsigned cell = (unsigned)cx * (unsigned)g + (unsigned)cy;
      atomicOr(&ws[ACC_WORDS + woff_of(s) + (cell >> 5)], 1u << (cell & 31u));
    }
  }

  // block-level min/max via LDS atomics, then one global atomic per size per block
  __shared__ int smin[NS], smax[NS];
  if (threadIdx.x < NS) { smin[threadIdx.x] = 0x7FFFFFFF; smax[threadIdx.x] = (int)0x80000000; }
  __syncthreads();
#pragma unroll
  for (int s = 0; s < NS; ++s) {
    atomicMin(&smin[s], lmin[s]);
    atomicMax(&smax[s], lmax[s]);
  }
  __syncthreads();
  if (threadIdx.x < NS) {
    atomicMin((int*)&ws[6u + threadIdx.x],  smin[threadIdx.x]);
    atomicMax((int*)&ws[12u + threadIdx.x], smax[threadIdx.x]);
  }
}

// ---------------- kernel 3: popcount bitmask, double-buffered async LDS pipeline ---
__global__ void mpc_count(unsigned* __restrict__ ws) {
  __shared__ unsigned buf[2][CHUNK_WORDS];
  const unsigned* __restrict__ mask = ws + ACC_WORDS;
  const unsigned tid = threadIdx.x;

  int lcnt[NS] = {0, 0, 0, 0, 0, 0};

  unsigned c = blockIdx.x;
  unsigned parity = 0u;
  if (c < NCHUNKS) issue_chunk(mask, c * CHUNK_WORDS, buf[0], tid);  // prologue

  for (; c < NCHUNKS; c += gridDim.x) {
    const unsigned nextc = c + gridDim.x;
    if (nextc < NCHUNKS) {
      issue_chunk(mask, nextc * CHUNK_WORDS, buf[parity ^ 1u], tid);  // prefetch next
      WAIT_ASYNC(4);   // oldest 4 (current chunk) done; next chunk stays in flight
    } else {
      WAIT_ASYNC(0);
    }
    __syncthreads();   // all waves' DMAs for chunk c visible in LDS

    const unsigned base = c * CHUNK_WORDS;
    const unsigned* row = buf[parity];
#pragma unroll
    for (unsigned k = 0; k < CHUNK_WORDS / 256u; ++k) {
      const unsigned w = tid + k * 256u;
      const unsigned gw = base + w;
      int s = (gw < 524288u) ? 0
            : (gw < 757362u) ? 1
            : (gw < 888434u) ? 2
            : (gw < 946660u) ? 3
            : (gw < 979428u) ? 4 : 5;   // zero-padded tail lands in 5, adds 0
      lcnt[s] += __popc(row[w]);
    }
    __syncthreads();   // before the buffer we just read gets rewritten
    parity ^= 1u;
  }

  __shared__ int scnt[NS];
  if (tid < NS) scnt[tid] = 0;
  __syncthreads();
#pragma unroll
  for (int s = 0; s < NS; ++s) atomicAdd(&scnt[s], lcnt[s]);
  __syncthreads();
  if (tid < NS) atomicAdd((int*)&ws[tid], scnt[tid]);
}

// ---------------- kernel 4: emit (3,6) as floats -----------------------------------
__global__ void mpc_finalize(const unsigned* __restrict__ ws, float* __restrict__ out) {
  int t = threadIdx.x;
  if (t < 18) {
    int r = t / 6, s = t % 6;
    int v = (r == 0) ? (int)ws[s] : (r == 1) ? (int)ws[6 + s] : (int)ws[12 + s];
    out[t] = (float)v;
  }
}

extern "C" void kernel_launch(void* const* d_in, const int* in_sizes, int n_in,
                              void* d_out, int out_size, void* d_ws, size_t ws_size,
                              hipStream_t stream) {
  const unsigned long long* pts64 = (const unsigned long long*)d_in[0];  // (P,2) f32 pairs
  const float* ps    = (const float*)d_in[1];
  const float* pcmin = (const float*)d_in[2];
  // d_in[3] (grid_sizes) is static in the reference; hardcoded above.
  unsigned* ws = (unsigned*)d_ws;
  float* out = (float*)d_out;
  const int npts = in_sizes[0] / 2;

  mpc_init<<<(TOTAL_WS_WORDS + 255u) / 256u, 256, 0, stream>>>(ws);
  mpc_scatter<<<2048, 256, 0, stream>>>(pts64, ps, pcmin, ws, npts);
  mpc_count<<<96, 256, 0, stream>>>(ws);   // ~2-3 chunks/block -> pipeline active
  mpc_finalize<<<1, 32, 0, stream>>>(ws, out);
}